// dotAttn_78606491452122
// MI455X (gfx1250) — compile-verified
//
#include <hip/hip_runtime.h>
#include <hip/hip_bf16.h>

#define B_ 64
#define T_ 2048
#define D_ 512
#define NEGV (-1e10f)

typedef __attribute__((ext_vector_type(2))) float v2f;
typedef __attribute__((ext_vector_type(8))) float v8f;

// ---------------- Kernel 1: q = query @ Wq + bq  ([B,D] x [D,D]) ----------------
__global__ __launch_bounds__(256)
void qproj_kernel(const float* __restrict__ query,
                  const float* __restrict__ Wq,
                  const float* __restrict__ bq,
                  float* __restrict__ qout) {
  __shared__ float qrow[D_];
  const int b = blockIdx.x;
  for (int i = threadIdx.x; i < D_; i += 256) qrow[i] = query[b * D_ + i];
  __syncthreads();
  for (int a = threadIdx.x; a < D_; a += 256) {
    float acc = bq[a];
#pragma unroll 8
    for (int k = 0; k < D_; ++k) acc = fmaf(qrow[k], Wq[k * D_ + a], acc);
    qout[b * D_ + a] = acc;
  }
}

// ---------------- Kernel 2: fused energy ----------------
// energy[b,t] = sum_a Wo[a] * tanh( (keys[b,t,:]@Wk)[a] + q[b,a] + cover[b,t]*Wc[a] )
// Workgroup: 8 waves = 2 M-slots (32 t-rows each) x 4 N-slots (128 a-cols each).
// GEMM via V_WMMA_F32_16X16X4_F32, accumulating f32, K = 512 in chunks of 32.
#define TB 64   // t rows per workgroup
#define KC 32   // K chunk staged in LDS
__global__ __launch_bounds__(256)
void energy_kernel(const float* __restrict__ keys,
                   const float* __restrict__ cover,
                   const float* __restrict__ Wk,
                   const float* __restrict__ Wc,
                   const float* __restrict__ Wo,
                   const float* __restrict__ qws,
                   float* __restrict__ energy) {
  __shared__ float lds_keys[TB * 33];  // stride 33 -> conflict-free A-frag reads
  __shared__ float lds_cover[TB];
  __shared__ float lds_energy[TB];

  const int tid  = threadIdx.x;
  const int b    = blockIdx.y;
  const int t0   = blockIdx.x * TB;
  const int wave = tid >> 5;
  const int lane = tid & 31;
  const int lm   = lane & 15;      // A: row M / B: col N / C: col N
  const int h    = lane >> 4;      // half-wave selector
  const int mslot = wave & 1;          // rows [0,32) or [32,64)
  const int nbase = (wave >> 1) * 128; // a-column base for this wave

  if (tid < TB) {
    lds_cover[tid]  = cover[b * T_ + t0 + tid];
    lds_energy[tid] = 0.f;
  }

  v8f acc[2][8] = {};  // 2 M-tiles x 8 N-tiles of 16x16 f32 accumulators

  const int row_ld = tid >> 3;  // 0..31
  const int quad   = tid & 7;   // 0..7

  for (int kc = 0; kc < D_ / KC; ++kc) {
    __syncthreads();
    // Cooperative stage of keys[b, t0..t0+63, kc*KC .. +KC) -> LDS.
    // 8 threads x float4 per row = one full 128B segment per row.
#pragma unroll
    for (int rr = 0; rr < 2; ++rr) {
      const int row = rr * 32 + row_ld;
      const float4 v = *(const float4*)(keys +
          ((size_t)(b * T_ + t0 + row)) * D_ + kc * KC + quad * 4);
      float* dst = &lds_keys[row * 33 + quad * 4];
      dst[0] = v.x; dst[1] = v.y; dst[2] = v.z; dst[3] = v.w;
    }
    __syncthreads();

#pragma unroll
    for (int kk = 0; kk < KC / 4; ++kk) {
      // A-fragment layout (16x4 f32): lane lm = M row, K = kbase + 2*h + {0,1}
      const int c = kk * 4 + 2 * h;
      v2f a0, a1;
      {
        const float* kr0 = &lds_keys[(mslot * 32 + lm) * 33 + c];
        const float* kr1 = &lds_keys[(mslot * 32 + 16 + lm) * 33 + c];
        a0.x = kr0[0]; a0.y = kr0[1];
        a1.x = kr1[0]; a1.y = kr1[1];
      }
      // B-fragment layout (4x16 f32): lane lm = N col, K = kbase + 2*h + vgpr
      const int krow = kc * KC + kk * 4 + 2 * h;
      const float* wkp = Wk + (size_t)krow * D_ + nbase + lm;
#pragma unroll
      for (int n = 0; n < 8; ++n) {
        v2f bf;
        bf.x = wkp[n * 16];
        bf.y = wkp[D_ + n * 16];
        acc[0][n] = __builtin_amdgcn_wmma_f32_16x16x4_f32(
            false, a0, false, bf, (short)0, acc[0][n], false, false);
        acc[1][n] = __builtin_amdgcn_wmma_f32_16x16x4_f32(
            false, a1, false, bf, (short)0, acc[1][n], false, false);
      }
    }
  }

  // Epilogue: per-lane params for its 8 a-columns
  float qv[8], wcv[8], wov[8];
#pragma unroll
  for (int n = 0; n < 8; ++n) {
    const int a = nbase + n * 16 + lm;
    qv[n]  = qws[b * D_ + a];
    wcv[n] = Wc[a];
    wov[n] = Wo[a];
  }
  __syncthreads();
  // C/D layout: VGPR r, lane -> row M = r + 8*h, col N = lm
#pragma unroll
  for (int m = 0; m < 2; ++m) {
#pragma unroll
    for (int r = 0; r < 8; ++r) {
      const int row = mslot * 32 + m * 16 + r + 8 * h;
      const float cov = lds_cover[row];
      float s = 0.f;
#pragma unroll
      for (int n = 0; n < 8; ++n) {
        const float v = acc[m][n][r] + qv[n] + cov * wcv[n];
        s += tanhf(v) * wov[n];
      }
      // butterfly over the 16 N-lanes of this half-wave
      s += __shfl_xor(s, 1);
      s += __shfl_xor(s, 2);
      s += __shfl_xor(s, 4);
      s += __shfl_xor(s, 8);
      if (lm == 0) atomicAdd(&lds_energy[row], s);  // ds_add_f32 across N-slot waves
    }
  }
  __syncthreads();
  if (tid < TB) energy[b * T_ + t0 + tid] = lds_energy[tid];
}

// ---------------- Kernel 3: masked softmax over T ----------------
__global__ __launch_bounds__(256)
void softmax_kernel(const float* __restrict__ energy,
                    const int* __restrict__ key_len,
                    float* __restrict__ attn) {
  __shared__ float red[8];
  const int b = blockIdx.x, tid = threadIdx.x;
  const int klen = key_len[b];
  float v[8];
  float mx = -3.4e38f;
#pragma unroll
  for (int i = 0; i < 8; ++i) {
    const int t = tid + i * 256;
    const float e = (t < klen) ? energy[b * T_ + t] : NEGV;  // SCALING == 1
    v[i] = e;
    mx = fmaxf(mx, e);
  }
  for (int off = 16; off > 0; off >>= 1) mx = fmaxf(mx, __shfl_xor(mx, off));
  if ((tid & 31) == 0) red[tid >> 5] = mx;
  __syncthreads();
  mx = red[0];
#pragma unroll
  for (int w = 1; w < 8; ++w) mx = fmaxf(mx, red[w]);

  float sum = 0.f;
#pragma unroll
  for (int i = 0; i < 8; ++i) { v[i] = __expf(v[i] - mx); sum += v[i]; }
  for (int off = 16; off > 0; off >>= 1) sum += __shfl_xor(sum, off);
  __syncthreads();
  if ((tid & 31) == 0) red[tid >> 5] = sum;
  __syncthreads();
  sum = 0.f;
#pragma unroll
  for (int w = 0; w < 8; ++w) sum += red[w];
  const float inv = 1.f / sum;
#pragma unroll
  for (int i = 0; i < 8; ++i) attn[b * T_ + tid + i * 256] = v[i] * inv;
}

// ---------------- Kernel 4: context partials (stream value @ 23.3 TB/s) ----------------
__global__ __launch_bounds__(128)
void context_partial_kernel(const float* __restrict__ attn,
                            const float* __restrict__ value,
                            float* __restrict__ part) {
  const int b  = blockIdx.z;
  const int d  = blockIdx.x * 128 + threadIdx.x;
  const int t0 = blockIdx.y * 256;
  const float* vp = value + ((size_t)(b * T_ + t0)) * D_ + d;
  const float* ap = attn + b * T_ + t0;
  float acc = 0.f;
  for (int i = 0; i < 256; ++i) acc = fmaf(ap[i], vp[(size_t)i * D_], acc);
  part[((b * 8 + blockIdx.y) * D_) + d] = acc;
}

// ---------------- Kernel 5: deterministic partial reduce ----------------
__global__ __launch_bounds__(512)
void context_reduce_kernel(const float* __restrict__ part, float* __restrict__ ctx) {
  const int b = blockIdx.x, d = threadIdx.x;
  float s = 0.f;
#pragma unroll
  for (int j = 0; j < 8; ++j) s += part[((b * 8 + j) * D_) + d];
  ctx[b * D_ + d] = s;
}

extern "C" void kernel_launch(void* const* d_in, const int* in_sizes, int n_in,
                              void* d_out, int out_size, void* d_ws, size_t ws_size,
                              hipStream_t stream) {
  const float* query   = (const float*)d_in[0];
  const float* keys    = (const float*)d_in[1];
  const float* value   = (const float*)d_in[2];
  const float* cover   = (const float*)d_in[3];
  const int*   key_len = (const int*)d_in[4];
  const float* Wq      = (const float*)d_in[5];
  const float* bq      = (const float*)d_in[6];
  const float* Wk      = (const float*)d_in[7];
  const float* Wc      = (const float*)d_in[8];
  const float* Wo      = (const float*)d_in[9];

  float* out_ctx  = (float*)d_out;           // [B, D]
  float* out_attn = (float*)d_out + B_ * D_; // [B, T]

  float* ws        = (float*)d_ws;
  float* q_ws      = ws;                 // B*D
  float* energy_ws = q_ws + B_ * D_;     // B*T
  float* part_ws   = energy_ws + B_ * T_;// B*8*D

  qproj_kernel<<<dim3(B_), dim3(256), 0, stream>>>(query, Wq, bq, q_ws);
  energy_kernel<<<dim3(T_ / TB, B_), dim3(256), 0, stream>>>(
      keys, cover, Wk, Wc, Wo, q_ws, energy_ws);
  softmax_kernel<<<dim3(B_), dim3(256), 0, stream>>>(energy_ws, key_len, out_attn);
  context_partial_kernel<<<dim3(D_ / 128, 8, B_), dim3(128), 0, stream>>>(
      out_attn, value, part_ws);
  context_reduce_kernel<<<dim3(B_), dim3(512), 0, stream>>>(part_ws, out_ctx);
}